// FastWeight_25838523253087
// MI455X (gfx1250) — compile-verified
//
#include <hip/hip_runtime.h>

// D = lam*A + eta * outer(h,h), batched [B,D,D], fp32, memory-bound streamer.
// One wave32 per 16x16 output tile; rank-1 update via V_WMMA_F32_16X16X4_F32.

typedef __attribute__((ext_vector_type(2))) float v2f;
typedef __attribute__((ext_vector_type(8))) float v8f;

#define LAM 0.95f
#define ETA 0.5f
#define DDIM 1024
#define TILES_PER_ROW (DDIM / 16)                    // 64
#define TILES_PER_MAT (TILES_PER_ROW * TILES_PER_ROW) // 4096
#define WAVES_PER_BLOCK 8

__global__ __launch_bounds__(256) void fastweight_wmma_kernel(
    const float* __restrict__ A,
    const float* __restrict__ h,
    float* __restrict__ out)
{
    const int lane = threadIdx.x & 31;
    const int wave = threadIdx.x >> 5;

    // Global tile index: one wave per 16x16 tile.
    const long long t = (long long)blockIdx.x * WAVES_PER_BLOCK + wave;

    const int b   = (int)(t >> 12);       // t / 4096
    const int rem = (int)(t & 4095);
    const int ti  = rem >> 6;             // tile row
    const int tj  = rem & 63;             // tile col

    const int i0 = ti * 16;
    const int j0 = tj * 16;
    const int n  = lane & 15;             // N (and M) position = lane % 16
    const int hi = lane >> 4;             // lane half selects M-offset 0/8 in C/D layout

    // Rank-1 operands. Fill ALL four K slices with 0.5-scaled values so the
    // result eta*h_i*h_j holds under any K<->(vgpr,lane-half) permutation:
    //   sum_k (0.5*eta*h_m)*(0.5*h_n) over k=0..3  =  eta*h_m*h_n
    const float hv_i = h[(size_t)b * DDIM + i0 + n];
    const float hv_j = h[(size_t)b * DDIM + j0 + n];

    v2f a_op, b_op;
    a_op.x = 0.5f * ETA * hv_i;
    a_op.y = a_op.x;
    b_op.x = 0.5f * hv_j;
    b_op.y = b_op.x;

    // C/D VGPR layout: vgpr v, lane L -> element (M = v + 8*(L>=16), N = L%16)
    const size_t matBase = (size_t)b * DDIM * DDIM;
    const size_t elt0    = matBase + (size_t)(i0 + hi * 8) * DDIM + (size_t)(j0 + n);
    const float* Ap = A + elt0;
    float*       Op = out + elt0;

    // C = lam * A_tile  (streaming, non-temporal: single-use data, keep L2 clean)
    v8f c;
#pragma unroll
    for (int v = 0; v < 8; ++v) {
        c[v] = LAM * __builtin_nontemporal_load(Ap + (size_t)v * DDIM);
    }

    // D = a_op x b_op + C   ->  v_wmma_f32_16x16x4_f32
    v8f d = __builtin_amdgcn_wmma_f32_16x16x4_f32(
        /*neg_a=*/false, a_op,
        /*neg_b=*/false, b_op,
        /*c_mod=*/(short)0, c,
        /*reuse_a=*/false, /*reuse_b=*/false);

#pragma unroll
    for (int v = 0; v < 8; ++v) {
        __builtin_nontemporal_store(d[v], Op + (size_t)v * DDIM);
    }
}

extern "C" void kernel_launch(void* const* d_in, const int* in_sizes, int n_in,
                              void* d_out, int out_size, void* d_ws, size_t ws_size,
                              hipStream_t stream) {
    const float* A = (const float*)d_in[0];
    const float* h = (const float*)d_in[1];
    float* out = (float*)d_out;

    const int Bn = in_sizes[1] / DDIM;                         // 128 batches
    const long long tiles = (long long)Bn * TILES_PER_MAT;     // 524288 tiles
    const int blocks = (int)(tiles / WAVES_PER_BLOCK);         // 65536 blocks

    fastweight_wmma_kernel<<<blocks, 256, 0, stream>>>(A, h, out);
}